// GNNModel_7155415515616
// MI455X (gfx1250) — compile-verified
//
#include <hip/hip_runtime.h>
#include <hip/hip_bf16.h>

typedef __attribute__((ext_vector_type(2))) float v2f;
typedef __attribute__((ext_vector_type(8))) float v8f;

// ---------------------------------------------------------------------------
// Degree / normalization
// ---------------------------------------------------------------------------
__global__ void zero_u32_kernel(unsigned* __restrict__ p, int n) {
    int i = blockIdx.x * blockDim.x + threadIdx.x;
    if (i < n) p[i] = 0u;
}

__global__ void count_deg_kernel(const int* __restrict__ dst,
                                 unsigned* __restrict__ deg, int e) {
    int i = blockIdx.x * blockDim.x + threadIdx.x;
    if (i < e) atomicAdd(&deg[dst[i]], 1u);
}

__global__ void calc_dinv_kernel(const unsigned* __restrict__ deg,
                                 float* __restrict__ dinv, int n) {
    int i = blockIdx.x * blockDim.x + threadIdx.x;
    if (i < n) dinv[i] = rsqrtf((float)(deg[i] + 1u));  // +1 for self loop
}

// ---------------------------------------------------------------------------
// H[n,64] = X[n,64] @ W[64,64] using V_WMMA_F32_16X16X4_F32.
// One 256-thread block = 8 waves; each wave owns one 16x16 output tile:
// 2 M-tiles (32 rows/block) x 4 N-tiles (64 cols). W staged in LDS.
// ---------------------------------------------------------------------------
__global__ __launch_bounds__(256) void gemm64_wmma_kernel(
    const float* __restrict__ X, const float* __restrict__ W,
    float* __restrict__ H, int n) {
    __shared__ float sW[64 * 64];

    // Stage W into LDS (4096 floats, 16 per thread as float4).
    {
        const float4* Wv = (const float4*)W;
        float4* sWv = (float4*)sW;
        #pragma unroll
        for (int i = 0; i < 4; ++i)
            sWv[threadIdx.x + i * 256] = Wv[threadIdx.x + i * 256];
    }
    __syncthreads();

    const int wave = threadIdx.x >> 5;        // 0..7
    const int lane = threadIdx.x & 31;
    const int mt   = wave >> 2;               // 0..1
    const int nt   = wave & 3;                // 0..3
    const int mbase = blockIdx.x * 32 + mt * 16;
    const int nbase = nt * 16;
    const int l15  = lane & 15;
    const int hi   = lane >> 4;               // 0 or 1

    int mrow = mbase + l15;
    if (mrow >= n) mrow = n - 1;              // clamp (EXEC must stay all-1s)
    const float* xrow = X + (size_t)mrow * 64;

    v8f c = {};
    #pragma unroll
    for (int k0 = 0; k0 < 64; k0 += 4) {
        const int ka = k0 + hi * 2;           // lanes 0-15: K={k0,k0+1}; 16-31: K={k0+2,k0+3}
        v2f a, b;
        a.x = xrow[ka + 0];
        a.y = xrow[ka + 1];
        b.x = sW[(ka + 0) * 64 + nbase + l15];
        b.y = sW[(ka + 1) * 64 + nbase + l15];
        // 8 args: (neg_a, A, neg_b, B, c_mod, C, reuse_a, reuse_b)
        c = __builtin_amdgcn_wmma_f32_16x16x4_f32(
                false, a, false, b, (short)0, c, false, false);
    }

    // C/D layout: lanes 0-15 VGPR r -> M=r; lanes 16-31 VGPR r -> M=r+8.
    #pragma unroll
    for (int r = 0; r < 8; ++r) {
        const int row = mbase + hi * 8 + r;
        if (row < n) H[(size_t)row * 64 + nbase + l15] = c[r];
    }
}

// ---------------------------------------------------------------------------
// Zero the accumulator (float4 stores).
// ---------------------------------------------------------------------------
__global__ void zero_f32x4_kernel(float4* __restrict__ p, int n4) {
    int i = blockIdx.x * blockDim.x + threadIdx.x;
    if (i < n4) p[i] = make_float4(0.f, 0.f, 0.f, 0.f);
}

// ---------------------------------------------------------------------------
// Edge aggregation: AGG[dst] += H[src] * dinv[src]*dinv[dst].
// 16 threads per edge, float4 gather + 4 native fp32 atomics.
// ---------------------------------------------------------------------------
__global__ __launch_bounds__(256) void aggregate_kernel(
    const float* __restrict__ H, const int* __restrict__ src,
    const int* __restrict__ dst, const float* __restrict__ dinv,
    float* __restrict__ AGG, int nE) {
    int t = blockIdx.x * 256 + threadIdx.x;
    int e = t >> 4;
    if (e >= nE) return;
    int f = (t & 15) << 2;
    int s = src[e];
    int d = dst[e];
    float norm = dinv[s] * dinv[d];
    float4 v = *(const float4*)(H + (size_t)s * 64 + f);
    float* p = AGG + (size_t)d * 64 + f;
    unsafeAtomicAdd(p + 0, v.x * norm);
    unsafeAtomicAdd(p + 1, v.y * norm);
    unsafeAtomicAdd(p + 2, v.z * norm);
    unsafeAtomicAdd(p + 3, v.w * norm);
}

// ---------------------------------------------------------------------------
// Finalize: AGG = relu(AGG + H*dinv^2 + b)  (self-loop term folded in).
// ---------------------------------------------------------------------------
__global__ __launch_bounds__(256) void finalize_kernel(
    float* __restrict__ AGG, const float* __restrict__ H,
    const float* __restrict__ dinv, const float* __restrict__ bias, int n) {
    int t = blockIdx.x * 256 + threadIdx.x;
    int node = t >> 4;
    if (node >= n) return;
    int f = (t & 15) << 2;
    float di = dinv[node];
    float selfw = di * di;
    float4 a  = *(const float4*)(AGG + (size_t)node * 64 + f);
    float4 h  = *(const float4*)(H   + (size_t)node * 64 + f);
    float4 bb = *(const float4*)(bias + f);
    float4 r;
    r.x = fmaxf(a.x + h.x * selfw + bb.x, 0.f);
    r.y = fmaxf(a.y + h.y * selfw + bb.y, 0.f);
    r.z = fmaxf(a.z + h.z * selfw + bb.z, 0.f);
    r.w = fmaxf(a.w + h.w * selfw + bb.w, 0.f);
    *(float4*)(AGG + (size_t)node * 64 + f) = r;
}

// ---------------------------------------------------------------------------
// Output head: out[n,2] = X[n,64] @ Wl[64,2] + bl. One thread per node.
// ---------------------------------------------------------------------------
__global__ __launch_bounds__(256) void linear_out_kernel(
    const float* __restrict__ X, const float* __restrict__ Wl,
    const float* __restrict__ bl, float* __restrict__ out, int n) {
    int i = blockIdx.x * blockDim.x + threadIdx.x;
    if (i >= n) return;
    const float* xr = X + (size_t)i * 64;
    float a0 = bl[0], a1 = bl[1];
    #pragma unroll 8
    for (int k = 0; k < 64; ++k) {
        float v = xr[k];
        a0 += v * Wl[2 * k + 0];
        a1 += v * Wl[2 * k + 1];
    }
    out[2 * i + 0] = a0;
    out[2 * i + 1] = a1;
}

// ---------------------------------------------------------------------------
// Host launcher
// ---------------------------------------------------------------------------
extern "C" void kernel_launch(void* const* d_in, const int* in_sizes, int n_in,
                              void* d_out, int out_size, void* d_ws, size_t ws_size,
                              hipStream_t stream) {
    const float* x  = (const float*)d_in[0];
    const int* ei   = (const int*)d_in[1];
    // d_in[2] edge_attr: unused (matches reference)
    const float* W1 = (const float*)d_in[3];
    const float* b1 = (const float*)d_in[4];
    const float* W2 = (const float*)d_in[5];
    const float* b2 = (const float*)d_in[6];
    const float* Wl = (const float*)d_in[7];
    const float* bl = (const float*)d_in[8];
    float* out = (float*)d_out;

    const int n = in_sizes[0] / 64;
    const int e = in_sizes[1] / 2;
    const int* src = ei;
    const int* dst = ei + e;

    char* ws = (char*)d_ws;
    const size_t NB = (size_t)n * 64 * sizeof(float);
    float*    H    = (float*)(ws);
    float*    AGG  = (float*)(ws + NB);
    float*    DINV = (float*)(ws + 2 * NB);
    unsigned* DEG  = (unsigned*)(ws + 2 * NB + (size_t)n * sizeof(float));

    const int nodeBlocks  = (n + 255) / 256;
    const int edgeBlocks  = (e + 255) / 256;
    const int gemmBlocks  = (n + 31) / 32;
    const int featThreads = n * 16;                 // 16 threads/node (float4)
    const int featBlocks  = (featThreads + 255) / 256;
    const int aggThreads  = e * 16;                 // 16 threads/edge
    const int aggBlocks   = (aggThreads + 255) / 256;
    const int zeroN4      = n * 16;                 // n*64/4 float4's
    const int zeroBlocks  = (zeroN4 + 255) / 256;

    // degrees + normalization (shared by both layers)
    zero_u32_kernel<<<nodeBlocks, 256, 0, stream>>>(DEG, n);
    count_deg_kernel<<<edgeBlocks, 256, 0, stream>>>(dst, DEG, e);
    calc_dinv_kernel<<<nodeBlocks, 256, 0, stream>>>(DEG, DINV, n);

    // ---- layer 1 ----
    gemm64_wmma_kernel<<<gemmBlocks, 256, 0, stream>>>(x, W1, H, n);
    zero_f32x4_kernel<<<zeroBlocks, 256, 0, stream>>>((float4*)AGG, zeroN4);
    aggregate_kernel<<<aggBlocks, 256, 0, stream>>>(H, src, dst, DINV, AGG, e);
    finalize_kernel<<<featBlocks, 256, 0, stream>>>(AGG, H, DINV, b1, n);
    // AGG now holds layer-1 activations

    // ---- layer 2 ----
    gemm64_wmma_kernel<<<gemmBlocks, 256, 0, stream>>>(AGG, W2, H, n);
    zero_f32x4_kernel<<<zeroBlocks, 256, 0, stream>>>((float4*)AGG, zeroN4);
    aggregate_kernel<<<aggBlocks, 256, 0, stream>>>(H, src, dst, DINV, AGG, e);
    finalize_kernel<<<featBlocks, 256, 0, stream>>>(AGG, H, DINV, b2, n);
    // AGG now holds layer-2 activations

    // ---- output head ----
    linear_out_kernel<<<nodeBlocks, 256, 0, stream>>>(AGG, Wl, bl, out, n);
}